// QNNLayer_52450140619031
// MI455X (gfx1250) — compile-verified
//
#include <hip/hip_runtime.h>

typedef float v2f __attribute__((ext_vector_type(2)));
typedef float v8f __attribute__((ext_vector_type(8)));

#define NQ      10
#define NL      3
#define NSTATE  1024   // 2^NQ

// Fused single-qubit gate G = Rot(phi,theta,omega) * RY(x) (2x2 complex).
// g = {00r,00i, 01r,01i, 10r,10i, 11r,11i}
__device__ __forceinline__ void fused_gate(float xv, float phi, float th,
                                           float om, float g[8])
{
    float c  = __cosf(0.5f * xv), s = __sinf(0.5f * xv);
    float ct = __cosf(0.5f * th), st = __sinf(0.5f * th);
    float ap = 0.5f * (phi + om), am = 0.5f * (phi - om);
    float ca = __cosf(ap), sa = __sinf(ap);
    float cb = __cosf(am), sb = __sinf(am);
    float m00r =  ca * ct, m00i = -sa * ct;
    float m01r = -cb * st, m01i = -sb * st;
    float m10r =  cb * st, m10i = -sb * st;
    float m11r =  ca * ct, m11i =  sa * ct;
    g[0] =  m00r * c + m01r * s;  g[1] =  m00i * c + m01i * s;   // G00
    g[2] = -m00r * s + m01r * c;  g[3] = -m00i * s + m01i * c;   // G01
    g[4] =  m10r * c + m11r * s;  g[5] =  m10i * c + m11i * s;   // G10
    g[6] = -m10r * s + m11r * c;  g[7] = -m10i * s + m11i * c;   // G11
}

// One wave32 per batch element; state (1024 complex f32) in LDS, ping-pong
// buffers. Two-qubit fused gates applied as replicated 8x8 real matrix x
// 8x16 quadruple columns via chained V_WMMA_F32_16X16X4_F32 (K=8).
__global__ __launch_bounds__(32)
void qnn_circuit_kernel(const float* __restrict__ x,
                        const float* __restrict__ wts,
                        float* __restrict__ out)
{
    __shared__ float2 bufA[NSTATE];
    __shared__ float2 bufB[NSTATE];

    const int lane = threadIdx.x;   // 0..31 (wave32)
    const int b    = blockIdx.x;    // batch element

    for (int j = 0; j < NSTATE / 32; ++j) {
        int i = j * 32 + lane;
        bufA[i] = make_float2(i == 0 ? 1.0f : 0.0f, 0.0f);
    }

    float2* cur = bufA;
    float2* nxt = bufB;

    const int n16 = lane & 15;        // column within WMMA (quadruple id low)
    const int kh  = lane >> 4;        // lane half
    // A-fragment row decomposition (rows replicated mod 8):
    const int p  = lane & 1;          // re/im row parity
    const int i0 = (lane >> 1) & 1;   // bit_qa of output component
    const int i1 = (lane >> 2) & 1;   // bit_qb of output component

    for (int l = 0; l < NL; ++l) {
        for (int gp = 0; gp < NQ / 2; ++gp) {
            const int qa = 2 * gp, qb = 2 * gp + 1;
            // Gb acts on qa (low bit of component), Ga on qb (high bit)
            float gb[8], ga[8];
            fused_gate(x[b * NQ + qa], wts[(l * NQ + qa) * 3 + 0],
                       wts[(l * NQ + qa) * 3 + 1], wts[(l * NQ + qa) * 3 + 2], gb);
            fused_gate(x[b * NQ + qb], wts[(l * NQ + qb) * 3 + 0],
                       wts[(l * NQ + qb) * 3 + 1], wts[(l * NQ + qb) * 3 + 2], ga);

            // Per-lane Kron entries. Gb column = kh; Ga column = k-step t.
            float gbr = i0 ? (kh ? gb[6] : gb[4]) : (kh ? gb[2] : gb[0]);
            float gbi = i0 ? (kh ? gb[7] : gb[5]) : (kh ? gb[3] : gb[1]);
            float ga0r = i1 ? ga[4] : ga[0], ga0i = i1 ? ga[5] : ga[1];
            float ga1r = i1 ? ga[6] : ga[2], ga1i = i1 ? ga[7] : ga[3];
            float c0r = ga0r * gbr - ga0i * gbi, c0i = ga0r * gbi + ga0i * gbr;
            float c1r = ga1r * gbr - ga1i * gbi, c1i = ga1r * gbi + ga1i * gbr;
            // Real expansion: s=0 -> (p? Ci : Cr), s=1 -> (p? Cr : -Ci)
            v2f af0, af1;
            af0.x = p ? c0i : c0r;  af0.y = p ? c0r : -c0i;
            af1.x = p ? c1i : c1r;  af1.y = p ? c1r : -c1i;

            const int maskA = (1 << qa) - 1;
            const int maskB = (1 << qb) - 1;
            const int hopA  = 1 << qa;
            const int hopB  = 1 << qb;

            // quadruple base index for this lane at iteration `it`
            auto baseIdx = [&](int it) {
                int t   = it * 16 + n16;
                int tm  = ((t  & ~maskA) << 1) | (t  & maskA);
                return    ((tm & ~maskB) << 1) | (tm & maskB);
            };

            // software pipeline: prefetch iteration 0
            int ib   = baseIdx(0);
            int la   = ib | (kh << qa);
            float2 f0 = cur[la];
            float2 f1 = cur[la | hopB];

            for (int it = 0; it < 16; ++it) {
                // prefetch next iteration's B fragments (wraps harmlessly)
                int ibn = baseIdx((it + 1) & 15);
                int lan = ibn | (kh << qa);
                float2 nf0 = cur[lan];
                float2 nf1 = cur[lan | hopB];

                v2f b0; b0.x = f0.x; b0.y = f0.y;
                v2f b1; b1.x = f1.x; b1.y = f1.y;
                v8f acc = {};
                acc = __builtin_amdgcn_wmma_f32_16x16x4_f32(
                    false, af0, false, b0, (short)0, acc, false, false);
                acc = __builtin_amdgcn_wmma_f32_16x16x4_f32(
                    false, af1, false, b1, (short)0, acc, false, false);

                // lanes<16 store amp00',amp01' (d0..d3);
                // lanes>=16 store amp10',amp11' (d4..d7) -- rows replicated.
                int sa = ib | (kh << qb);
                int sb = sa | hopA;
                float2 o0, o1;
                o0.x = kh ? acc[4] : acc[0];  o0.y = kh ? acc[5] : acc[1];
                o1.x = kh ? acc[6] : acc[2];  o1.y = kh ? acc[7] : acc[3];
                cur[sa] = o0;
                cur[sb] = o1;

                ib = ibn; f0 = nf0; f1 = nf1;
            }
        }

        // ---- CNOT ring: fixed basis permutation, scatter cur -> nxt
        for (int j = 0; j < NSTATE / 32; ++j) {
            int i  = j * 32 + lane;
            int bb = i;
#pragma unroll
            for (int q2 = 0; q2 < NQ; ++q2) {
                int t = (q2 + 1) % NQ;
                bb ^= ((bb >> q2) & 1) << t;   // flip target if control set
            }
            nxt[bb] = cur[i];
        }
        { float2* tmp = cur; cur = nxt; nxt = tmp; }
    }

    // ---- readout: <Z_q> = sum_i |amp_i|^2 * (1 - 2*bit_q(i))
    float acc[NQ];
#pragma unroll
    for (int q = 0; q < NQ; ++q) acc[q] = 0.0f;
    for (int j = 0; j < NSTATE / 32; ++j) {
        int i = j * 32 + lane;
        float2 a = cur[i];
        float pr = a.x * a.x + a.y * a.y;
#pragma unroll
        for (int q = 0; q < NQ; ++q)
            acc[q] += ((i >> q) & 1) ? -pr : pr;
    }
#pragma unroll
    for (int q = 0; q < NQ; ++q) {
        float v = acc[q];
        for (int off = 16; off > 0; off >>= 1)
            v += __shfl_xor(v, off, 32);
        acc[q] = v;
    }
    if (lane == 0) {
#pragma unroll
        for (int q = 0; q < NQ; ++q)
            out[b * NQ + q] = acc[q];
    }
}

extern "C" void kernel_launch(void* const* d_in, const int* in_sizes, int n_in,
                              void* d_out, int out_size, void* d_ws, size_t ws_size,
                              hipStream_t stream)
{
    const float* x   = (const float*)d_in[0];   // (4096, 10) f32
    const float* wts = (const float*)d_in[1];   // (3, 10, 3) f32
    float* out = (float*)d_out;                 // (4096, 10) f32
    int B = in_sizes[0] / NQ;
    qnn_circuit_kernel<<<B, 32, 0, stream>>>(x, wts, out);
}